// LocallyConnected3DFlipout_14817637171813
// MI455X (gfx1250) — compile-verified
//
#include <hip/hip_runtime.h>
#include <cstddef>

// Problem constants (match reference)
#define NB   32          // batch
#define DD   30          // input spatial
#define CC   4           // in-channels
#define KSZ  3           // kernel size
#define ODIM 28          // output spatial (valid)
#define PPOS (ODIM*ODIM*ODIM)   // 21952 positions
#define KK   (KSZ*KSZ*KSZ*CC)   // 108 taps
#define FF   4           // filters

typedef float v2f __attribute__((ext_vector_type(2)));
typedef float v8f __attribute__((ext_vector_type(8)));
typedef int   v4i __attribute__((vector_size(16)));   // matches builtin param
typedef __attribute__((address_space(1))) v4i gv4i;   // global int4
typedef __attribute__((address_space(3))) v4i lv4i;   // LDS int4

#if defined(__gfx1250__) && __has_builtin(__builtin_amdgcn_global_load_async_to_lds_b128)
#define HAVE_ASYNC_LDS 1
#endif

__device__ __forceinline__ float softplus_f(float v) {
  // log1p(exp(v)) with overflow guard
  return (v > 20.0f) ? v : log1pf(__expf(v));
}

// One wave (32 threads) per output position p.
// WMMA mapping per position: D[32xF] += A[32xK] * B[KxF]
//   - M = batch (two 16-row tiles), K stepped 4 at a time via
//     V_WMMA_F32_16X16X4_F32, N columns 0..3 = filters.
//   - B tiles live in LDS pre-padded to 16 columns (cols 4..15 == 0) so the
//     inner loop is branch-free straight-line ds_load + v_wmma.
//   - k is stored in permuted order k' = (i*9+j*3+l)*4 + c for BOTH patches
//     and weights (contraction is permutation-invariant); this makes the
//     4 channel stores per tap two 128B-apart pairs -> ds_store_2addr.
__global__ __launch_bounds__(32)
void lc3d_flipout_wmma_f32(const float* __restrict__ x,
                           const float* __restrict__ kloc,
                           const float* __restrict__ krho,
                           const float* __restrict__ bias,
                           const float* __restrict__ eps,
                           const float* __restrict__ sign_in,
                           const float* __restrict__ sign_out,
                           float* __restrict__ out) {
  // LDS: patches stored [k'][b] so A-fragment reads are conflict-free.
  __shared__ float pX [KK * 32];   // patches(x)                 13.8 KB
  __shared__ float pXS[KK * 32];   // patches(x * sign_in)       13.8 KB
  __shared__ float w1p[KK * 16];   // kernel_loc[p], N-padded     6.9 KB
  __shared__ float w2p[KK * 16];   // softplus(rho)*eps, padded   6.9 KB

  const int p    = blockIdx.x;          // 0..PPOS-1
  const int d2   = p % ODIM;
  const int d1   = (p / ODIM) % ODIM;
  const int d0   = p / (ODIM * ODIM);
  const int lane = threadIdx.x;         // 0..31

  // ---- Stage patches: lane = batch index b ------------------------------
  // x[b, d0+i, d1+j, d2+l, c]; (l,c) contiguous -> 16B-aligned float4 loads.
  // Store at k' = (i*9 + j*3 + l)*4 + c  (channel innermost).
  {
    const int b = lane;
    #pragma unroll
    for (int i = 0; i < KSZ; ++i) {
      #pragma unroll
      for (int j = 0; j < KSZ; ++j) {
        const size_t base =
            ((((size_t)b * DD + (d0 + i)) * DD + (d1 + j)) * DD + d2) * CC;
        #pragma unroll
        for (int l = 0; l < KSZ; ++l) {
          const float4 vx = *(const float4*)(x       + base + (size_t)l * CC);
          const float4 vs = *(const float4*)(sign_in + base + (size_t)l * CC);
          const int kb = (i * 9 + j * 3 + l) * 4;   // k' base for c=0..3
          const float xs[4] = {vx.x, vx.y, vx.z, vx.w};
          const float ss[4] = {vs.x, vs.y, vs.z, vs.w};
          #pragma unroll
          for (int c = 0; c < CC; ++c) {
            pX [(kb + c) * 32 + b] = xs[c];
            pXS[(kb + c) * 32 + b] = xs[c] * ss[c];
          }
        }
      }
    }
  }

  // ---- Stage weights (N-padded to 16 cols, permuted k order) ------------
  // Global k order: k = c*27 + koff  ->  LDS row k' = koff*4 + c.
  {
    const float* locp = kloc + (size_t)p * (KK * FF);
    const float* rhop = krho + (size_t)p * (KK * FF);
    const float* epsp = eps  + (size_t)p * (KK * FF);
    const float4 z4 = make_float4(0.f, 0.f, 0.f, 0.f);
    for (int k = lane; k < KK; k += 32) {
      const int c    = k / 27;
      const int koff = k - c * 27;
      const int kp   = koff * 4 + c;          // permuted LDS row

      // kernel_loc tile: pure copy -> CDNA5 async global->LDS (ASYNCcnt),
      // no VGPR round-trip. Fallback: synchronous copy.
#ifdef HAVE_ASYNC_LDS
      __builtin_amdgcn_global_load_async_to_lds_b128(
          (gv4i*)(locp + (size_t)k * 4),
          (lv4i*)(w1p + kp * 16),
          /*offset=*/0, /*cpol=*/0);
#else
      *(float4*)(w1p + kp * 16) = *(const float4*)(locp + (size_t)k * 4);
#endif
      *(float4*)(w1p + kp * 16 + 4)  = z4;
      *(float4*)(w1p + kp * 16 + 8)  = z4;
      *(float4*)(w1p + kp * 16 + 12) = z4;

      const float4 r4 = *(const float4*)(rhop + (size_t)k * 4);
      const float4 e4 = *(const float4*)(epsp + (size_t)k * 4);
      float4 s4;
      s4.x = softplus_f(r4.x) * e4.x;
      s4.y = softplus_f(r4.y) * e4.y;
      s4.z = softplus_f(r4.z) * e4.z;
      s4.w = softplus_f(r4.w) * e4.w;
      *(float4*)(w2p + kp * 16)      = s4;
      *(float4*)(w2p + kp * 16 + 4)  = z4;
      *(float4*)(w2p + kp * 16 + 8)  = z4;
      *(float4*)(w2p + kp * 16 + 12) = z4;
    }
  }

#ifdef HAVE_ASYNC_LDS
#if __has_builtin(__builtin_amdgcn_s_wait_asynccnt)
  __builtin_amdgcn_s_wait_asynccnt(0);
#else
  asm volatile("s_wait_asynccnt 0x0" ::: "memory");
#endif
#endif
  __syncthreads();

  // ---- WMMA main loop (fully unrolled, branch-free) ---------------------
  // A (16x4 f32, 2 VGPRs): lane L -> row M = L%16; lanes<16 hold K={0,1},
  //   lanes>=16 hold K={2,3} (VGPR index selects within the pair).
  // B (4x16 f32, 2 VGPRs): lane L -> col N = L%16; same K striping as A.
  const int n   = lane & 15;   // N column (filter for n<4)
  const int hi  = lane >> 4;   // 0: K-pair {0,1} / 1: K-pair {2,3}
  const int kr  = hi * 2;
  const int m0  = lane & 15;   // batch row, tile0; tile1 adds 16

  v8f accM0 = {}; v8f accM1 = {};  // mean path, batch tiles 0/1
  v8f accN0 = {}; v8f accN1 = {};  // noise path, batch tiles 0/1

  #pragma unroll
  for (int kb = 0; kb < KK; kb += 4) {
    const int k0 = kb + kr;
    v2f a0, a1, s0, s1, bm, bn;
    a0[0] = pX [(k0 + 0) * 32 + m0];
    a0[1] = pX [(k0 + 1) * 32 + m0];
    a1[0] = pX [(k0 + 0) * 32 + m0 + 16];
    a1[1] = pX [(k0 + 1) * 32 + m0 + 16];
    s0[0] = pXS[(k0 + 0) * 32 + m0];
    s0[1] = pXS[(k0 + 1) * 32 + m0];
    s1[0] = pXS[(k0 + 0) * 32 + m0 + 16];
    s1[1] = pXS[(k0 + 1) * 32 + m0 + 16];
    bm[0] = w1p[(k0 + 0) * 16 + n];
    bm[1] = w1p[(k0 + 1) * 16 + n];
    bn[0] = w2p[(k0 + 0) * 16 + n];
    bn[1] = w2p[(k0 + 1) * 16 + n];

    accM0 = __builtin_amdgcn_wmma_f32_16x16x4_f32(false, a0, false, bm,
                                                  (short)0, accM0, false, false);
    accM1 = __builtin_amdgcn_wmma_f32_16x16x4_f32(false, a1, false, bm,
                                                  (short)0, accM1, false, false);
    accN0 = __builtin_amdgcn_wmma_f32_16x16x4_f32(false, s0, false, bn,
                                                  (short)0, accN0, false, false);
    accN1 = __builtin_amdgcn_wmma_f32_16x16x4_f32(false, s1, false, bn,
                                                  (short)0, accN1, false, false);
  }

  // ---- Epilogue ---------------------------------------------------------
  // C/D layout: lane L, vgpr v -> M = v + 8*(L>=16), N = L%16.
  if (n < FF) {
    const float bf = bias[n];
    #pragma unroll
    for (int v = 0; v < 8; ++v) {
      const int b0 = v + hi * 8;         // batch row, tile 0
      const size_t o0 = ((size_t)b0 * PPOS + p) * FF + n;
      out[o0] = accM0[v] + sign_out[o0] * accN0[v] + bf;

      const int b1 = b0 + 16;            // batch row, tile 1
      const size_t o1 = ((size_t)b1 * PPOS + p) * FF + n;
      out[o1] = accM1[v] + sign_out[o1] * accN1[v] + bf;
    }
  }
}

extern "C" void kernel_launch(void* const* d_in, const int* in_sizes, int n_in,
                              void* d_out, int out_size, void* d_ws, size_t ws_size,
                              hipStream_t stream) {
  // setup_inputs() order: x, kernel_loc, kernel_rho, bias, eps, sign_in, sign_out
  const float* x        = (const float*)d_in[0];
  const float* kloc     = (const float*)d_in[1];
  const float* krho     = (const float*)d_in[2];
  const float* bias     = (const float*)d_in[3];
  const float* eps      = (const float*)d_in[4];
  const float* sign_in  = (const float*)d_in[5];
  const float* sign_out = (const float*)d_in[6];
  float* out            = (float*)d_out;

  dim3 grid(PPOS);   // one wave32 workgroup per output position
  dim3 block(32);
  hipLaunchKernelGGL(lc3d_flipout_wmma_f32, grid, block, 0, stream,
                     x, kloc, krho, bias, eps, sign_in, sign_out, out);
}